// GaussianAugment_17944373362935
// MI455X (gfx1250) — compile-verified
//
#include <hip/hip_runtime.h>
#include <math.h>

typedef __attribute__((ext_vector_type(2))) float v2f;
typedef __attribute__((ext_vector_type(8))) float v8f;

#define G_DIM 24
#define NN 576             // G*G
#define HD 64
#define NP 577             // N + PREFIX
#define BH 96              // 8*12
#define TILES_PER_BH 36    // 576/16
#define NTILES (BH * TILES_PER_BH)  // 3456
#define COEF 576.0f
#define EPS 0.1f

__global__ __launch_bounds__(256) void gauss_aug_kernel(
    const float* __restrict__ q,   // (8,12,577,64)
    const float* __restrict__ Wv,  // (64,2) row-major
    const float* __restrict__ bv,  // (2,)
    const float* __restrict__ Wa,  // (64,1)
    const float* __restrict__ ba,  // (1,)
    float* __restrict__ out)       // (8,12,577,577)
{
  const float B_SHIFT = 6.3544897055851564f; // ln(575)
  const int blk = blockIdx.x;
  const int tid = threadIdx.x;

  if (blk >= NTILES) {
    // zero the PREFIX pad row (n = 0) for this (b,h)
    int bh = blk - NTILES;
    float* o = out + (size_t)bh * NP * NP;
    for (int c = tid; c < NP; c += 256) o[c] = 0.0f;
    return;
  }

  __shared__ float ldsW[HD * 3];       // merged weights: [k][{Wv0,Wv1,Wa}]
  __shared__ float ldsX[16 * 3];       // raw WMMA logits per row
  __shared__ float ldsP[16 * 4];       // r0, r1, alpha per row
  __shared__ float ldsF[16 * G_DIM];   // exp table, i-dimension
  __shared__ float ldsG[16 * G_DIM];   // exp table, j-dimension

  const int bh   = blk / TILES_PER_BH;
  const int tile = blk % TILES_PER_BH;
  const int row0 = tile * 16;          // first n-row of this 16-row tile

  // ---- stage merged weight matrix in LDS (cols 0,1 = W_var, col 2 = W_alpha) ----
  for (int i = tid; i < HD; i += 256) {
    ldsW[i * 3 + 0] = Wv[i * 2 + 0];
    ldsW[i * 3 + 1] = Wv[i * 2 + 1];
    ldsW[i * 3 + 2] = Wa[i];
  }
  __syncthreads();

  // ---- Wave 0: projection GEMM via V_WMMA_F32_16X16X4_F32, K=64 in 16 steps ----
  if (tid < 32) {
    const int l    = tid;
    const int mrow = l & 15;     // M index (also B's N column for this lane)
    const int hi   = l >> 4;     // 0: K+{0,1}, 1: K+{2,3}
    const float* qrow = q + ((size_t)bh * NP + 1 + row0 + mrow) * HD;

    // branchless B-fragment addressing: clamp column, zero-scale cols >= 3
    const float colScale = (mrow < 3) ? 1.0f : 0.0f;
    const int   colIdx   = (mrow < 3) ? mrow : 0;

    v8f acc = {};
#pragma unroll
    for (int kt = 0; kt < 16; ++kt) {
      const int kk = 4 * kt + 2 * hi;
      v2f a = *(const v2f*)(qrow + kk);           // global_load_b64
      v2f b;
      b.x = ldsW[kk * 3 + colIdx] * colScale;     // ds_load_b32 + cndmask-free scale
      b.y = ldsW[(kk + 1) * 3 + colIdx] * colScale;
      acc = __builtin_amdgcn_wmma_f32_16x16x4_f32(false, a, false, b,
                                                  (short)0, acc, false, false);
    }
    // D layout: VGPR r holds row M = r + 8*hi, column N = lane&15
    if (mrow < 3) {
#pragma unroll
      for (int r = 0; r < 8; ++r) {
        ldsX[(r + 8 * hi) * 3 + mrow] = acc[r];
      }
    }
  }
  __syncthreads();

  // ---- per-row nonlinearity -> r0 = -0.5/(var0+eps), r1, alpha ----
  if (tid < 16) {
    float x0 = ldsX[tid * 3 + 0] + bv[0] - B_SHIFT;
    float x1 = ldsX[tid * 3 + 1] + bv[1] - B_SHIFT;
    float x2 = ldsX[tid * 3 + 2] + ba[0];
    float var0  = COEF / (1.0f + __expf(-x0));
    float var1  = COEF / (1.0f + __expf(-x1));
    float alpha = (x2 > 30.0f) ? x2 : __logf(1.0f + __expf(x2));
    ldsP[tid * 4 + 0] = -0.5f / (var0 + EPS);
    ldsP[tid * 4 + 1] = -0.5f / (var1 + EPS);
    ldsP[tid * 4 + 2] = alpha;
  }
  __syncthreads();

  // ---- separable exp tables: 24 + 24 entries per row instead of 576 exps ----
  for (int idx = tid; idx < 16 * 2 * G_DIM; idx += 256) {
    const int row = idx / (2 * G_DIM);
    const int rem = idx % (2 * G_DIM);
    const int n   = row0 + row;
    if (rem < G_DIM) {
      float di = (float)(n / G_DIM - rem);
      ldsF[row * G_DIM + rem] = __expf(ldsP[row * 4 + 0] * di * di);
    } else {
      float dj = (float)(n % G_DIM - (rem - G_DIM));
      ldsG[row * G_DIM + (rem - G_DIM)] = __expf(ldsP[row * 4 + 1] * dj * dj);
    }
  }
  __syncthreads();

  // ---- stream 16 x 577 coalesced output stores (col 0 = pad = 0) ----
  float* obase = out + (size_t)bh * NP * NP + (size_t)(1 + row0) * NP;
  for (int idx = tid; idx < 16 * NP; idx += 256) {
    const int row = idx / NP;
    const int c   = idx - row * NP;
    float v = 0.0f;
    if (c > 0) {
      const int m  = c - 1;
      const int im = m / G_DIM;
      const int jm = m - im * G_DIM;
      v = ldsP[row * 4 + 2] * ldsF[row * G_DIM + im] * ldsG[row * G_DIM + jm];
    }
    obase[(size_t)row * NP + c] = v;
  }
}

extern "C" void kernel_launch(void* const* d_in, const int* in_sizes, int n_in,
                              void* d_out, int out_size, void* d_ws, size_t ws_size,
                              hipStream_t stream) {
  const float* q  = (const float*)d_in[0];
  const float* Wv = (const float*)d_in[1];
  const float* bv = (const float*)d_in[2];
  const float* Wa = (const float*)d_in[3];
  const float* ba = (const float*)d_in[4];
  float* out = (float*)d_out;
  (void)in_sizes; (void)n_in; (void)out_size; (void)d_ws; (void)ws_size;

  gauss_aug_kernel<<<NTILES + BH, 256, 0, stream>>>(q, Wv, bv, Wa, ba, out);
}